// Attention_39676907886014
// MI455X (gfx1250) — compile-verified
//
#include <hip/hip_runtime.h>
#include <hip/hip_bf16.h>

typedef __attribute__((ext_vector_type(2))) float v2f;
typedef __attribute__((ext_vector_type(8))) float v8f;

#define WMMA_F32_16x16x4(A, B, C) \
    __builtin_amdgcn_wmma_f32_16x16x4_f32(false, (A), false, (B), (short)0, (C), false, false)

constexpr int BATCH = 2;
constexpr int HEADS = 16;
constexpr int SQ    = 2048;
constexpr int SKV   = 2048;
constexpr int DHEAD = 64;
constexpr int QROWS = 32;              // query rows per wave (2 x 16-row tiles)
constexpr int WAVES_PER_BLOCK = 8;     // 256 threads, wave32

// exchange value with the other 16-lane half of the wave (SWAPX16)
__device__ __forceinline__ float swap_half16(float x) {
    int i = __builtin_amdgcn_ds_swizzle(__float_as_int(x), 0x401f); // xor=0x10,and=0x1f
    return __int_as_float(i);
}

__global__ __launch_bounds__(256, 1) void
fattn_f32_wmma(const float* __restrict__ Q, const float* __restrict__ K,
               const float* __restrict__ V, const float* __restrict__ ISF,
               float* __restrict__ O)
{
    const int lane = threadIdx.x & 31;
    const int wave = threadIdx.x >> 5;
    const int task = blockIdx.x * WAVES_PER_BLOCK + wave;   // [0, BH * SQ/32)
    const int bh   = task >> 6;                             // / (SQ/32)
    const int q0   = (task & 63) << 5;                      // 32-row query block base
    const int l16  = lane & 15;                             // n-index (q) / m-index
    const int h2   = (lane >> 4) << 1;                      // 0 or 2 (A/B frag k offset)
    const int h8   = (lane >> 4) << 3;                      // 0 or 8 (C-layout m offset)

    const float rscale = 1.0f / ISF[bh];

    const float* qp = Q + ((size_t)bh * SQ + q0) * DHEAD;
    const float* kp = K + (size_t)bh * SKV * DHEAD;
    const float* vp = V + (size_t)bh * SKV * DHEAD;

    // ---- preload Q as B-fragments of Q^T for both 16-row tiles, scale folded --
    // tile t, step s, reg a: Q[q = q0 + 16t + l16][d = 4s + h2 + a] * rscale
    v2f qf[2][16];
    #pragma unroll
    for (int t = 0; t < 2; ++t) {
        const float* qr = qp + (size_t)(16 * t + l16) * DHEAD + h2;
        #pragma unroll
        for (int s = 0; s < 16; ++s) {
            qf[t][s].x = qr[4 * s + 0] * rscale;
            qf[t][s].y = qr[4 * s + 1] * rscale;
        }
    }

    // O^T accumulators: acc[t][mt] holds d = 16*mt + (v + h8), q = 16t + l16
    v8f acc[2][4];
    #pragma unroll
    for (int t = 0; t < 2; ++t)
        #pragma unroll
        for (int mt = 0; mt < 4; ++mt) acc[t][mt] = (v8f)0.0f;

    float rowM[2] = {-INFINITY, -INFINITY};  // running max (per q = l16, dup halves)
    float rowL[2] = {0.0f, 0.0f};            // running sumexp

    for (int cb = 0; cb < SKV; cb += 16) {
        // ---- gemm1: S^T[kv][q] = sum_d K[kv][d] * Q[q][d] (scaled), 2 tiles --
        // K A-frag (step s, reg a): K[cb + l16][4s + h2 + a]; shared by tiles.
        v8f st[2];
        st[0] = (v8f)0.0f;
        st[1] = (v8f)0.0f;
        const float* kr = kp + (size_t)(cb + l16) * DHEAD + h2;
        #pragma unroll
        for (int s = 0; s < 16; ++s) {
            v2f kf;
            kf.x = kr[4 * s + 0];
            kf.y = kr[4 * s + 1];
            st[0] = WMMA_F32_16x16x4(kf, qf[0][s], st[0]);   // two independent
            st[1] = WMMA_F32_16x16x4(kf, qf[1][s], st[1]);   // accumulation chains
        }
        // st[t][v] = S_scaled[q = q0 + 16t + l16][kv = cb + v + h8]

        // ---- online softmax over this 16-wide kv chunk, per tile --------------
        float alpha[2];
        #pragma unroll
        for (int t = 0; t < 2; ++t) {
            float mloc = st[t][0];
            #pragma unroll
            for (int v = 1; v < 8; ++v) mloc = fmaxf(mloc, st[t][v]);
            mloc = fmaxf(mloc, swap_half16(mloc));          // combine both halves
            const float mnew = fmaxf(rowM[t], mloc);
            alpha[t] = __expf(rowM[t] - mnew);              // exp(-inf - x)=0 first it
            float psum = 0.0f;
            #pragma unroll
            for (int v = 0; v < 8; ++v) {
                st[t][v] = __expf(st[t][v] - mnew);         // p values, stay in place
                psum += st[t][v];
            }
            psum += swap_half16(psum);
            rowL[t] = rowL[t] * alpha[t] + psum;
            rowM[t] = mnew;
            #pragma unroll
            for (int mt = 0; mt < 4; ++mt) acc[t][mt] *= alpha[t];
        }

        // ---- gemm2: O^T[d][q] += sum_kv V[kv][d] * P^T[kv][q] ----------------
        // K-dim mapping kv = s + 4*k_local => B-frag is {st[t][s], st[t][s+4]}
        // in-lane; V A-frags shared by both tiles.
        #pragma unroll
        for (int s = 0; s < 4; ++s) {
            const float* vr0 = vp + (size_t)(cb + s + h8) * DHEAD + l16;  // a=0 row
            const float* vr1 = vr0 + 4 * DHEAD;                           // a=1 row
            v2f af[4];
            #pragma unroll
            for (int mt = 0; mt < 4; ++mt) {          // V^T A-frag: d = 16*mt + l16
                af[mt].x = vr0[16 * mt];
                af[mt].y = vr1[16 * mt];
            }
            #pragma unroll
            for (int t = 0; t < 2; ++t) {
                v2f bf;
                bf.x = st[t][s];
                bf.y = st[t][s + 4];
                #pragma unroll
                for (int mt = 0; mt < 4; ++mt)
                    acc[t][mt] = WMMA_F32_16x16x4(af[mt], bf, acc[t][mt]);
            }
        }
    }

    // ---- normalize and store: O[q = q0 + 16t + l16][d = 16*mt + v + h8] ------
    #pragma unroll
    for (int t = 0; t < 2; ++t) {
        const float linv = 1.0f / rowL[t];
        float* op = O + ((size_t)bh * SQ + q0 + 16 * t + l16) * DHEAD + h8;
        #pragma unroll
        for (int mt = 0; mt < 4; ++mt) {
            #pragma unroll
            for (int v = 0; v < 8; ++v)
                op[16 * mt + v] = acc[t][mt][v] * linv;
        }
    }
}

extern "C" void kernel_launch(void* const* d_in, const int* in_sizes, int n_in,
                              void* d_out, int out_size, void* d_ws, size_t ws_size,
                              hipStream_t stream) {
    const float* Q   = (const float*)d_in[0];
    const float* K   = (const float*)d_in[1];
    const float* V   = (const float*)d_in[2];
    const float* ISF = (const float*)d_in[3];
    float* Out = (float*)d_out;
    (void)in_sizes; (void)n_in; (void)out_size; (void)d_ws; (void)ws_size;

    const int tasks  = BATCH * HEADS * (SQ / QROWS);          // 2048 waves
    const int blocks = tasks / WAVES_PER_BLOCK;               // 256
    fattn_f32_wmma<<<blocks, WAVES_PER_BLOCK * 32, 0, stream>>>(Q, K, V, ISF, Out);
}